// LengthRegulator_49400713839081
// MI455X (gfx1250) — compile-verified
//
#include <hip/hip_runtime.h>
#include <hip/hip_bf16.h>
#include <math.h>

// Problem constants (match reference)
#define BB   32
#define LL   768
#define DD   256
#define FF   256
#define TT   3072
#define ROWS (BB * LL)      // 24576 GEMM rows
#define KK   (3 * DD)       // 768  im2col K
#define LNEPS 1e-5f

typedef __attribute__((ext_vector_type(2))) float v2f;
typedef __attribute__((ext_vector_type(8))) float v8f;

// -------------------------------------------------------------------------
// Fused conv1d(K=3,pad=1) as im2col GEMM via V_WMMA_F32_16X16X4_F32,
// + bias + ReLU + LayerNorm.  Block = 512 threads (16 waves, wave32).
// Each block computes a 16-row x 256-col output tile; wave w owns N-tile w.
// in : [B*L, 256] row-major, W : [768, 256] row-major (HIO conv weights).
// -------------------------------------------------------------------------
#define A_STRIDE 772   // 768 + 4 -> rotates LDS banks per row
#define H_STRIDE 260

__global__ __launch_bounds__(512) void conv_relu_ln_wmma(
    const float* __restrict__ in, const float* __restrict__ W,
    const float* __restrict__ bias, const float* __restrict__ gamma,
    const float* __restrict__ beta, float* __restrict__ out)
{
    __shared__ float sA[16 * A_STRIDE];          // 49408 B; reused for h tile

    const int tid   = threadIdx.x;
    const int lane  = tid & 31;
    const int wv    = tid >> 5;                  // 0..15 -> N tile
    const int half  = lane >> 4;                 // 0/1: K sub-select
    const int m     = lane & 15;                 // A row / B,D column

    const int r0 = blockIdx.x * 16;              // first global row of tile
    const int b  = r0 / LL;                      // L%16==0 -> tile never spans batches
    const int l0 = r0 % LL;

    // ---- stage padded im2col A tile into LDS: sA[m][k], k = tap*256 + c ----
    for (int e = tid; e < 16 * KK; e += 512) {
        const int mr  = e / KK;
        const int k   = e % KK;
        const int tap = k >> 8;                  // 0,1,2
        const int c   = k & 255;
        const int l   = l0 + mr + tap - 1;       // conv pad = 1
        float v = 0.0f;
        if (l >= 0 && l < LL) v = in[((size_t)(b * LL + l)) * DD + c];
        sA[mr * A_STRIDE + k] = v;
    }
    __syncthreads();

    // ---- K loop: 192 fp32 WMMAs per wave ----
    const int n0 = wv * 16;
    v8f acc = {};
    for (int k0 = 0; k0 < KK; k0 += 4) {
        const int ka = k0 + 2 * half;            // this half-wave's K pair
        v2f a, bm;
        a.x  = sA[m * A_STRIDE + ka + 0];        // A[m][ka],  A[m][ka+1]
        a.y  = sA[m * A_STRIDE + ka + 1];
        bm.x = W[(size_t)(ka + 0) * FF + n0 + m];// B[ka][n],  B[ka+1][n]
        bm.y = W[(size_t)(ka + 1) * FF + n0 + m];
        acc = __builtin_amdgcn_wmma_f32_16x16x4_f32(
            false, a, false, bm, (short)0, acc, false, false);
    }
    __syncthreads();                             // all A reads done -> reuse LDS

    // ---- bias + ReLU, park D tile in LDS (VGPR i: rows i / i+8) ----
    float* sH = sA;                              // [16][H_STRIDE]
    {
        const float bv = bias[n0 + m];
        #pragma unroll
        for (int i = 0; i < 8; ++i) {
            float v = acc[i] + bv;
            v = v > 0.0f ? v : 0.0f;
            sH[(i + 8 * half) * H_STRIDE + n0 + m] = v;
        }
    }
    __syncthreads();

    // ---- LayerNorm: wave w normalizes row w (256 elems, 8 per lane) ----
    float vals[8], s = 0.0f, ss = 0.0f;
    #pragma unroll
    for (int j = 0; j < 8; ++j) {
        float v = sH[wv * H_STRIDE + lane + 32 * j];
        vals[j] = v; s += v; ss += v * v;
    }
    #pragma unroll
    for (int off = 16; off >= 1; off >>= 1) {
        s  += __shfl_xor(s,  off, 32);
        ss += __shfl_xor(ss, off, 32);
    }
    const float mu  = s  * (1.0f / FF);
    const float var = ss * (1.0f / FF) - mu * mu;
    const float rs  = rsqrtf(var + LNEPS);
    const size_t rowBase = (size_t)(r0 + wv) * FF;
    #pragma unroll
    for (int j = 0; j < 8; ++j) {
        const int c = lane + 32 * j;
        out[rowBase + c] = (vals[j] - mu) * rs * gamma[c] + beta[c];
    }
}

// -------------------------------------------------------------------------
// durations = exp(h @ lin_w + lin_b); wave-per-row reduction.
// -------------------------------------------------------------------------
__global__ __launch_bounds__(256) void linear_exp_kernel(
    const float* __restrict__ h, const float* __restrict__ w,
    const float* __restrict__ bsc, float* __restrict__ dur)
{
    const int lane = threadIdx.x & 31;
    const int r    = blockIdx.x * 8 + (threadIdx.x >> 5);
    float s = 0.0f;
    #pragma unroll
    for (int j = 0; j < 8; ++j) {
        const int c = lane + 32 * j;
        s += h[(size_t)r * FF + c] * w[c];
    }
    #pragma unroll
    for (int off = 16; off >= 1; off >>= 1) s += __shfl_xor(s, off, 32);
    if (lane == 0) dur[r] = expf(s + bsc[0]);
}

// -------------------------------------------------------------------------
// dpo = float(target); cum = inclusive scan of target per batch row.
// One block per batch, 768 threads, Hillis-Steele in LDS.
// -------------------------------------------------------------------------
__global__ __launch_bounds__(768) void scan_kernel(
    const int* __restrict__ target, float* __restrict__ dpo,
    int* __restrict__ cum)
{
    __shared__ int s[LL];
    const int b = blockIdx.x, t = threadIdx.x;
    const int v = target[b * LL + t];
    s[t] = v;
    __syncthreads();
    for (int off = 1; off < LL; off <<= 1) {
        int add = (t >= off) ? s[t - off] : 0;
        __syncthreads();
        s[t] += add;
        __syncthreads();
    }
    cum[b * LL + t] = s[t];
    dpo[b * LL + t] = (float)v;   // round(target * 1.0)
}

// -------------------------------------------------------------------------
// Length regulator gather: one block per (b,t); searchsorted-right + copy.
// -------------------------------------------------------------------------
__global__ __launch_bounds__(256) void gather_kernel(
    const float* __restrict__ x, const int* __restrict__ cum,
    float* __restrict__ out)
{
    const int bt = blockIdx.x;
    const int b  = bt / TT;
    const int t  = bt % TT;
    const int* c = cum + b * LL;

    // first index where cum[idx] > t  (side='right')
    int lo = 0, hi = LL;
    while (lo < hi) {
        const int mid = (lo + hi) >> 1;
        if (c[mid] <= t) lo = mid + 1; else hi = mid;
    }
    const int   ic  = lo < (LL - 1) ? lo : (LL - 1);
    const float msk = (t < c[LL - 1]) ? 1.0f : 0.0f;

    out[(size_t)bt * DD + threadIdx.x] =
        x[((size_t)(b * LL + ic)) * DD + threadIdx.x] * msk;
}

// -------------------------------------------------------------------------
extern "C" void kernel_launch(void* const* d_in, const int* in_sizes, int n_in,
                              void* d_out, int out_size, void* d_ws, size_t ws_size,
                              hipStream_t stream) {
    (void)in_sizes; (void)n_in; (void)out_size; (void)ws_size;
    const float* x        = (const float*)d_in[0];
    const int*   target   = (const int*)  d_in[1];
    /* d_in[2] = mel_max_length scalar (compile-time TT here) */
    const float* conv1_w  = (const float*)d_in[3];
    const float* conv1_b  = (const float*)d_in[4];
    const float* ln1_g    = (const float*)d_in[5];
    const float* ln1_b    = (const float*)d_in[6];
    const float* conv2_w  = (const float*)d_in[7];
    const float* conv2_b  = (const float*)d_in[8];
    const float* ln2_g    = (const float*)d_in[9];
    const float* ln2_b    = (const float*)d_in[10];
    const float* lin_w    = (const float*)d_in[11];
    const float* lin_b    = (const float*)d_in[12];

    float* out_lr   = (float*)d_out;                        // [B,T,D]
    float* out_dpo  = out_lr + (size_t)BB * TT * DD;        // [B,L]
    float* out_dur  = out_dpo + (size_t)BB * LL;            // [B,L]

    char*  ws  = (char*)d_ws;
    float* h1  = (float*)ws;                                // 24576*256 f32
    float* h2  = h1 + (size_t)ROWS * FF;
    int*   cum = (int*)(h2 + (size_t)ROWS * FF);            // 24576 i32

    // conv1 -> relu -> LN
    conv_relu_ln_wmma<<<ROWS / 16, 512, 0, stream>>>(x,  conv1_w, conv1_b, ln1_g, ln1_b, h1);
    // conv2 -> relu -> LN
    conv_relu_ln_wmma<<<ROWS / 16, 512, 0, stream>>>(h1, conv2_w, conv2_b, ln2_g, ln2_b, h2);
    // linear + exp
    linear_exp_kernel<<<ROWS / 8, 256, 0, stream>>>(h2, lin_w, lin_b, out_dur);
    // dpo + cumsum
    scan_kernel<<<BB, LL, 0, stream>>>(target, out_dpo, cum);
    // length-regulator gather
    gather_kernel<<<BB * TT, DD, 0, stream>>>(x, cum, out_lr);
}